// SiameseSpectralModel_74036646249258
// MI455X (gfx1250) — compile-verified
//
#include <hip/hip_runtime.h>
#include <hip/hip_bf16.h>
#include <math.h>
#include <stdint.h>

typedef __attribute__((ext_vector_type(2))) float v2f;
typedef __attribute__((ext_vector_type(8))) float v8f;

#define NUM_BINS    100000
#define GROUPS      3333
#define GROUP_SIZE  30
#define OUT_PER_GRP 3
#define BINNER_OUT  9999          // GROUPS * OUT_PER_GRP
#define KPAD1       10000         // BINNER_OUT padded (multiple of KC)
#define BATCH       256
#define PEAKS       512
#define MROWS       512           // 2 * BATCH (both spectra batched)
#define HID         1000
#define EOUT        500
#define KC          40            // K-chunk staged into LDS (divides 10000 and 1000)

// ---------------------------------------------------------------------------
// Fill h0[MROWS][KPAD1] with grouped-linear bias (bg) and zero K-padding.
// ---------------------------------------------------------------------------
__global__ __launch_bounds__(256) void init_h0_kernel(float* __restrict__ h0,
                                                      const float* __restrict__ bg) {
    int idx = blockIdx.x * blockDim.x + threadIdx.x;
    if (idx >= MROWS * KPAD1) return;
    int col = idx % KPAD1;
    h0[idx] = (col < BINNER_OUT) ? bg[col] : 0.0f;
}

// ---------------------------------------------------------------------------
// Fused histogram + grouped tiny linear:
// peak with bin idx = g*30+i adds v*Wg[g][i][o] to h0[row][3g+o].
// ---------------------------------------------------------------------------
__global__ __launch_bounds__(PEAKS) void scatter_peaks_kernel(
    const float* __restrict__ mz1, const float* __restrict__ in1,
    const float* __restrict__ mz2, const float* __restrict__ in2,
    const float* __restrict__ Wg,  float* __restrict__ h0) {
    int row  = blockIdx.x;        // 0..511: rows 0..255 spectrum1, 256..511 spectrum2
    int spec = row >> 8;
    int b    = row & 255;
    const float* mz = spec ? mz2 : mz1;
    const float* iv = spec ? in2 : in1;
    int p = threadIdx.x;
    float m = mz[b * PEAKS + p];
    float v = iv[b * PEAKS + p];
    if (!(m >= 0.0f && m < 1000.0f)) return;
    int idx = (int)floorf(m / 0.01f);
    if (idx < 0) idx = 0;
    if (idx > NUM_BINS - 1) idx = NUM_BINS - 1;
    if (idx >= GROUPS * GROUP_SIZE) return;   // bins 99990..99999 are sliced off
    int g = idx / GROUP_SIZE;
    int i = idx - g * GROUP_SIZE;
    const float* w = Wg + ((size_t)g * GROUP_SIZE + i) * OUT_PER_GRP;
    float* dst = h0 + (size_t)row * KPAD1 + g * OUT_PER_GRP;
    atomicAdd(dst + 0, v * w[0]);
    atomicAdd(dst + 1, v * w[1]);
    atomicAdd(dst + 2, v * w[2]);
}

// ---------------------------------------------------------------------------
// fp32 WMMA GEMM with async LDS staging of B:
//   C[M][N] = relu?(A[M][K] @ W[K][N] + bias[N])
// A: lda == Kpad (multiple of KC), padding region zero => unguarded A loads.
// B: KC x 64 chunk cooperatively staged via global_load_async_to_lds_b128,
//    double-buffered; OOB rows/cols address-clamped (finite data; products
//    with zero-padded A vanish, invalid columns are never stored).
// Wave tile 16(M) x 64(N); 4 waves stack M => block tile 64x64.
// grid = (ceil(N/64), M/64), block = 128 threads.
// ---------------------------------------------------------------------------
template<bool RELU>
__global__ __launch_bounds__(128) void gemm_wmma_kernel(
    const float* __restrict__ A, int lda, int K, int Kpad,
    const float* __restrict__ W, const float* __restrict__ bias,
    float* __restrict__ C, int N) {
    const int lane = threadIdx.x & 31;
    const int wave = threadIdx.x >> 5;
    const int lo   = lane & 15;     // M-row (A frag), N-col (B/C/D frag)
    const int hi   = lane >> 4;     // K-pair selector / M+8 half

    const int rowTile = blockIdx.y * 64 + wave * 16;
    const int colBase = blockIdx.x * 64;

    __shared__ float ldsB[2][KC * 64];          // 2 x 10 KB double buffer
    const unsigned ldsB0 = (unsigned)(uintptr_t)&ldsB[0][0];
    const uint64_t wbase = (uint64_t)(uintptr_t)W;

    // ---- async stage of one KC x 64 B-chunk (5 x b128 per lane) ----
    auto stage = [&](int buf, int kk) {
        #pragma unroll
        for (int s = 0; s < 5; ++s) {
            int l    = (int)threadIdx.x + 128 * s;   // 0..639 segments of 16B
            int row  = l >> 4;                       // 0..KC-1
            int segc = (l & 15) << 2;                // col offset 0,4,..,60
            int gr   = kk + row;      if (gr > K - 1) gr = K - 1;       // clamp K
            int gc   = colBase + segc; if (gc + 4 > N) gc = N - 4;      // clamp N
            unsigned goff  = ((unsigned)gr * (unsigned)N + (unsigned)gc) * 4u;
            unsigned laddr = ldsB0 + (unsigned)(buf * (KC * 64 * 4) + (row * 64 + segc) * 4);
            asm volatile("global_load_async_to_lds_b128 %0, %1, %2"
                         :: "v"(laddr), "v"(goff), "s"(wbase) : "memory");
        }
    };

    // Clamped column indices for bias/store
    const int col0 = colBase + lo;
    const int col1 = col0 + 16, col2 = col0 + 32, col3 = col0 + 48;
    const int c0 = (col0 < N) ? col0 : (N - 1);
    const int c1 = (col1 < N) ? col1 : (N - 1);
    const int c2 = (col2 < N) ? col2 : (N - 1);
    const int c3 = (col3 < N) ? col3 : (N - 1);

    v8f acc0 = {}, acc1 = {}, acc2 = {}, acc3 = {};

    const unsigned aIdx = (unsigned)(rowTile + lo) * (unsigned)lda + (unsigned)(2 * hi);
    const int nchunks = Kpad / KC;

    stage(0, 0);
    for (int c = 0; c < nchunks; ++c) {
        const int cur = c & 1;
        if (c + 1 < nchunks) {
            stage(cur ^ 1, (c + 1) * KC);
            asm volatile("s_wait_asynccnt %0" :: "i"(5) : "memory");  // cur chunk done
        } else {
            asm volatile("s_wait_asynccnt %0" :: "i"(0) : "memory");
        }
        __syncthreads();                      // all waves' staged portions visible

        const float* bb = &ldsB[cur][2 * hi * 64];   // row k-local = 2*hi
        #pragma unroll
        for (int step = 0; step < KC / 4; ++step) {
            v2f a = *(const v2f*)(A + aIdx + (unsigned)(c * KC + step * 4));
            const float* br = bb + step * 4 * 64;    // immediate LDS offsets
            v2f b0 = { br[lo],      br[lo + 64] };   // B[k][col], B[k+1][col]
            v2f b1 = { br[lo + 16], br[lo + 80] };
            v2f b2 = { br[lo + 32], br[lo + 96] };
            v2f b3 = { br[lo + 48], br[lo + 112] };
            acc0 = __builtin_amdgcn_wmma_f32_16x16x4_f32(false, a, false, b0, (short)0, acc0, false, false);
            acc1 = __builtin_amdgcn_wmma_f32_16x16x4_f32(false, a, false, b1, (short)0, acc1, false, false);
            acc2 = __builtin_amdgcn_wmma_f32_16x16x4_f32(false, a, false, b2, (short)0, acc2, false, false);
            acc3 = __builtin_amdgcn_wmma_f32_16x16x4_f32(false, a, false, b3, (short)0, acc3, false, false);
        }
        __syncthreads();                      // reads done before buffer reuse
    }

    // Store: D VGPR r holds C[rowTile + r + 8*hi][colBase + t*16 + lo]
    const float bv0 = bias[c0], bv1 = bias[c1], bv2 = bias[c2], bv3 = bias[c3];
    #pragma unroll
    for (int r = 0; r < 8; ++r) {
        const int row = rowTile + r + 8 * hi;
        float v0 = acc0[r] + bv0;
        float v1 = acc1[r] + bv1;
        float v2 = acc2[r] + bv2;
        float v3 = acc3[r] + bv3;
        if (RELU) {
            v0 = fmaxf(v0, 0.0f); v1 = fmaxf(v1, 0.0f);
            v2 = fmaxf(v2, 0.0f); v3 = fmaxf(v3, 0.0f);
        }
        float* crow = C + (size_t)row * N;
        if (col0 < N) crow[col0] = v0;
        if (col1 < N) crow[col1] = v1;
        if (col2 < N) crow[col2] = v2;
        if (col3 < N) crow[col3] = v3;
    }
}

// ---------------------------------------------------------------------------
// Cosine similarity between rows b and b+256 of e[512][EOUT].
// ---------------------------------------------------------------------------
__global__ __launch_bounds__(256) void cosine_kernel(const float* __restrict__ e,
                                                     float* __restrict__ out) {
    const int b = blockIdx.x;
    const float* e1 = e + (size_t)b * EOUT;
    const float* e2 = e + (size_t)(b + BATCH) * EOUT;
    float dot = 0.0f, n1 = 0.0f, n2 = 0.0f;
    for (int i = threadIdx.x; i < EOUT; i += 256) {
        float x = e1[i], y = e2[i];
        dot += x * y; n1 += x * x; n2 += y * y;
    }
    __shared__ float sd[256], s1[256], s2[256];
    sd[threadIdx.x] = dot; s1[threadIdx.x] = n1; s2[threadIdx.x] = n2;
    __syncthreads();
    for (int s = 128; s > 0; s >>= 1) {
        if (threadIdx.x < s) {
            sd[threadIdx.x] += sd[threadIdx.x + s];
            s1[threadIdx.x] += s1[threadIdx.x + s];
            s2[threadIdx.x] += s2[threadIdx.x + s];
        }
        __syncthreads();
    }
    if (threadIdx.x == 0) {
        float nn1 = fmaxf(sqrtf(s1[0]), 1e-6f);
        float nn2 = fmaxf(sqrtf(s2[0]), 1e-6f);
        out[b] = sd[0] / (nn1 * nn2);
    }
}

// ---------------------------------------------------------------------------
extern "C" void kernel_launch(void* const* d_in, const int* in_sizes, int n_in,
                              void* d_out, int out_size, void* d_ws, size_t ws_size,
                              hipStream_t stream) {
    (void)in_sizes; (void)n_in; (void)out_size; (void)ws_size;
    const float* mz1 = (const float*)d_in[0];
    const float* in1 = (const float*)d_in[1];
    const float* mz2 = (const float*)d_in[2];
    const float* in2 = (const float*)d_in[3];
    const float* Wg  = (const float*)d_in[4];
    const float* bg  = (const float*)d_in[5];
    const float* W1  = (const float*)d_in[6];
    const float* b1  = (const float*)d_in[7];
    const float* W2  = (const float*)d_in[8];
    const float* b2  = (const float*)d_in[9];
    const float* W3  = (const float*)d_in[10];
    const float* b3  = (const float*)d_in[11];
    const float* Wo  = (const float*)d_in[12];
    const float* bo  = (const float*)d_in[13];
    float* out = (float*)d_out;

    // Workspace layout (floats): h0 | h1 | h2 | h3 | e   (~27.6 MB total)
    float* h0 = (float*)d_ws;                       // [512][10000]
    float* h1 = h0 + (size_t)MROWS * KPAD1;         // [512][1000]
    float* h2 = h1 + (size_t)MROWS * HID;           // [512][1000]
    float* h3 = h2 + (size_t)MROWS * HID;           // [512][1000]
    float* ev = h3 + (size_t)MROWS * HID;           // [512][500]

    // 1) bias-init + zero-pad h0
    init_h0_kernel<<<(MROWS * KPAD1 + 255) / 256, 256, 0, stream>>>(h0, bg);
    // 2) fused histogram + grouped linear (both spectra)
    scatter_peaks_kernel<<<MROWS, PEAKS, 0, stream>>>(mz1, in1, mz2, in2, Wg, h0);
    // 3) MLP via fp32 WMMA GEMMs with async LDS staging, M = 512
    gemm_wmma_kernel<true ><<<dim3((HID  + 63) / 64, MROWS / 64), 128, 0, stream>>>(h0, KPAD1, BINNER_OUT, KPAD1, W1, b1, h1, HID);
    gemm_wmma_kernel<true ><<<dim3((HID  + 63) / 64, MROWS / 64), 128, 0, stream>>>(h1, HID,   HID,        HID,   W2, b2, h2, HID);
    gemm_wmma_kernel<true ><<<dim3((HID  + 63) / 64, MROWS / 64), 128, 0, stream>>>(h2, HID,   HID,        HID,   W3, b3, h3, HID);
    gemm_wmma_kernel<false><<<dim3((EOUT + 63) / 64, MROWS / 64), 128, 0, stream>>>(h3, HID,   HID,        HID,   Wo, bo, ev, EOUT);
    // 4) cosine similarity
    cosine_kernel<<<BATCH, 256, 0, stream>>>(ev, out);
}